// EnhancedGNN_83571473645710
// MI455X (gfx1250) — compile-verified
//
#include <hip/hip_runtime.h>
#include <hip/hip_bf16.h>
#include <math.h>

// ---------------------------------------------------------------------------
// EnhancedGNN for MI455X (gfx1250, wave32).
// Dense node transforms via V_WMMA_F32_16X16X4_F32, compile-time K/N so the
// inner loop is pure load+WMMA (no runtime guards). One wave = 16x64 strip.
// K-tail and C-store edge handling are branchless (clamp + mask / uniform
// fast path) to avoid exec-mask churn. Edge gather/scatter via coalesced
// grid-stride kernels + float atomics.
// ---------------------------------------------------------------------------

typedef __attribute__((ext_vector_type(2))) float v2f;
typedef __attribute__((ext_vector_type(8))) float v8f;

#define DD 128   // hidden dim
#define HH 4     // heads
#define HD 512   // HH*DD

// ---------------- small helpers ----------------
__device__ __forceinline__ unsigned fenc(float f) {
  unsigned u = __float_as_uint(f);
  return (u & 0x80000000u) ? ~u : (u | 0x80000000u);   // monotonic order-preserving
}
__device__ __forceinline__ float fdec(unsigned u) {
  u = (u & 0x80000000u) ? (u & 0x7fffffffu) : ~u;
  return __uint_as_float(u);
}

__global__ void fill_f32(float* __restrict__ p, float v, int n) {
  int i = blockIdx.x * blockDim.x + threadIdx.x;
  int st = gridDim.x * blockDim.x;
  for (; i < n; i += st) p[i] = v;
}
__global__ void fill_u32(unsigned* __restrict__ p, unsigned v, int n) {
  int i = blockIdx.x * blockDim.x + threadIdx.x;
  int st = gridDim.x * blockDim.x;
  for (; i < n; i += st) p[i] = v;
}

// ---------------- WMMA fp32 GEMM: C[M,N] = (A (+A2)) @ B + bias ----------------
// A: M x K row-major (A2 optional fused elementwise add), B: K x N row-major,
// bias per column (nullable). K, N compile-time. One wave -> 16 x (16*NT) strip.
template <int K, int N, bool HASA2, int NT>
__global__ __launch_bounds__(256) void gemm_wmma(const float* __restrict__ A,
                                                 const float* __restrict__ A2,
                                                 const float* __restrict__ B,
                                                 const float* __restrict__ bias,
                                                 float* __restrict__ C, int M) {
  constexpr int TN = 16 * NT;
  const int tilesN = N / TN;
  const int wave = (blockIdx.x * blockDim.x + threadIdx.x) >> 5;
  const int lane = threadIdx.x & 31;
  const int half = lane >> 4;       // A/B frag: half 0 -> K=k,k+1 ; half 1 -> K=k+2,k+3
  const int r    = lane & 15;
  const int tm = wave / tilesN;
  const int tn = wave - tm * tilesN;
  if (tm * 16 >= M) return;

  int arow = tm * 16 + r;
  if (arow >= M) arow = M - 1;      // safe dup; M is 16-aligned in this workload
  const int col0 = tn * TN + r;

  v8f acc[NT];
#pragma unroll
  for (int nt = 0; nt < NT; ++nt) {
    const float bv = bias ? bias[col0 + nt * 16] : 0.0f;
#pragma unroll
    for (int i = 0; i < 8; ++i) acc[nt][i] = bv;
  }

  const float* __restrict__ Abase = A + (size_t)arow * K;   // row start (valid)
  const float* __restrict__ Ap = Abase + 2 * half;
  const float* A2base = nullptr;
  const float* A2p = nullptr;
  if constexpr (HASA2) {
    A2base = A2 + (size_t)arow * K;
    A2p = A2base + 2 * half;
  }
  const float* __restrict__ Bp = B + (size_t)(2 * half) * N + col0;

  constexpr int KB = K & ~3;        // full k=4 steps
#pragma unroll
  for (int k = 0; k < KB; k += 4) {
    v2f a = *(const v2f*)(Ap + k);
    if constexpr (HASA2) {
      v2f a2 = *(const v2f*)(A2p + k);
      a[0] += a2[0];
      a[1] += a2[1];
    }
#pragma unroll
    for (int nt = 0; nt < NT; ++nt) {
      v2f b;
      b[0] = Bp[(size_t)k * N + nt * 16];         // constexpr offsets -> IOFFSET
      b[1] = Bp[(size_t)(k + 1) * N + nt * 16];
      acc[nt] = __builtin_amdgcn_wmma_f32_16x16x4_f32(false, a, false, b,
                                                      (short)0, acc[nt], false, false);
    }
  }
  if constexpr ((K & 3) != 0) {
    // Branchless compile-time tail (only the K=10 path): clamp index + 0/1 mask,
    // so loads are unconditional (no exec-mask churn).
    const int ka = KB + 2 * half;
    const float m0 = (ka < K) ? 1.0f : 0.0f;
    const float m1 = (ka + 1 < K) ? 1.0f : 0.0f;
    const int i0 = (ka < K) ? ka : 0;
    const int i1 = (ka + 1 < K) ? (ka + 1) : 0;
    v2f a;
    a[0] = Abase[i0] * m0;
    a[1] = Abase[i1] * m1;
    if constexpr (HASA2) {
      a[0] += A2base[i0] * m0;
      a[1] += A2base[i1] * m1;
    }
    const float* __restrict__ B0 = B + (size_t)i0 * N + col0;
    const float* __restrict__ B1 = B + (size_t)i1 * N + col0;
#pragma unroll
    for (int nt = 0; nt < NT; ++nt) {
      v2f b;
      b[0] = B0[nt * 16] * m0;
      b[1] = B1[nt * 16] * m1;
      acc[nt] = __builtin_amdgcn_wmma_f32_16x16x4_f32(false, a, false, b,
                                                      (short)0, acc[nt], false, false);
    }
  }

  const int crow = tm * 16 + half * 8;  // lanes 0-15 -> rows 0..7, 16-31 -> 8..15
  float* __restrict__ Cp = C + (size_t)crow * N + col0;
  if (tm * 16 + 16 <= M) {
    // wave-uniform fast path: unguarded, fully coalesced stores
#pragma unroll
    for (int i = 0; i < 8; ++i)
#pragma unroll
      for (int nt = 0; nt < NT; ++nt) Cp[(size_t)i * N + nt * 16] = acc[nt][i];
  } else {
#pragma unroll
    for (int i = 0; i < 8; ++i) {
      if (crow + i < M) {
#pragma unroll
        for (int nt = 0; nt < NT; ++nt) Cp[(size_t)i * N + nt * 16] = acc[nt][i];
      }
    }
  }
}

template <int K, int N, bool HASA2>
static inline void launch_gemm_t(const float* A, const float* A2, const float* B,
                                 const float* bias, float* C, int M, hipStream_t s) {
  constexpr int NT = 4;                       // 16x64 strip per wave
  const long long waves = (long long)((M + 15) / 16) * (N / (16 * NT));
  const int blocks = (int)((waves * 32 + 255) / 256);
  gemm_wmma<K, N, HASA2, NT><<<blocks, 256, 0, s>>>(A, A2, B, bias, C, M);
}

// ---------------- BatchNorm: column stats (sum, sumsq) ----------------
__global__ void col_stats(const float* __restrict__ X, int M, int N,
                          float* __restrict__ acc /* [2*N], pre-zeroed */) {
  const int col = blockIdx.x * blockDim.x + threadIdx.x;
  if (col >= N) return;
  float s = 0.f, s2 = 0.f;
  for (int row = blockIdx.y; row < M; row += gridDim.y) {
    const float v = X[(long long)row * N + col];
    s += v; s2 += v * v;
  }
  atomicAdd(&acc[col], s);
  atomicAdd(&acc[N + col], s2);
}

__global__ void finalize_stats(const float* __restrict__ acc, int N, float invM,
                               float* __restrict__ mr /* [2*N]: mean, rstd */) {
  const int c = blockIdx.x * blockDim.x + threadIdx.x;
  if (c >= N) return;
  const float m = acc[c] * invM;
  const float v = acc[N + c] * invM - m * m;
  mr[c]     = m;
  mr[N + c] = rsqrtf(v + 1e-5f);
}

__global__ void bn_apply(const float* __restrict__ X, const float* __restrict__ mr,
                         const float* __restrict__ g, const float* __restrict__ b,
                         float* __restrict__ Y, int M, int N, int relu) {
  int i = blockIdx.x * blockDim.x + threadIdx.x;
  const int tot = M * N;
  const int st = gridDim.x * blockDim.x;
  for (; i < tot; i += st) {
    const int c = i % N;
    float y = (X[i] - mr[c]) * mr[N + c] * g[c] + b[c];
    if (relu) y = fmaxf(y, 0.f);
    Y[i] = y;
  }
}

// ---------------- GINE message + scatter: agg[dst] += relu(x[src] + ea@We + be) ----------------
__global__ void gine_edges(const float* __restrict__ x0, const int* __restrict__ ei,
                           const float* __restrict__ ea, const float* __restrict__ We,
                           const float* __restrict__ be, float* __restrict__ agg, int E) {
  int i = blockIdx.x * blockDim.x + threadIdx.x;
  const int tot = E * DD;
  const int st = gridDim.x * blockDim.x;
  for (; i < tot; i += st) {
    const int e = i >> 7;
    const int d = i & 127;
    const int s = ei[e];
    const int t = ei[E + e];
    float ep = be[d];
#pragma unroll
    for (int k = 0; k < 5; ++k) ep += ea[e * 5 + k] * We[k * DD + d];
    float m = x0[(long long)s * DD + d] + ep;
    m = fmaxf(m, 0.f);
    atomicAdd(&agg[(long long)t * DD + d], m);
  }
}

// ---------------- skip mix: out = sigmoid(skip)*a + (1-s)*relu(b + bias) ----------------
__global__ void skip_mix(const float* __restrict__ a, const float* __restrict__ b,
                         const float* __restrict__ bias, const float* __restrict__ skip,
                         float* __restrict__ out, int M, int N) {
  const float s = 1.f / (1.f + __expf(-skip[0]));
  int i = blockIdx.x * blockDim.x + threadIdx.x;
  const int tot = M * N;
  const int st = gridDim.x * blockDim.x;
  for (; i < tot; i += st) {
    const int c = i % N;
    const float xb = b[i] + (bias ? bias[c] : 0.f);
    out[i] = s * a[i] + (1.f - s) * fmaxf(xb, 0.f);
  }
}

// ---------------- GATv2 attention logits: one wave32 per edge ----------------
__global__ void gat_logits(const float* __restrict__ xl, const float* __restrict__ xr,
                           const int* __restrict__ ei, const float* __restrict__ ea,
                           const float* __restrict__ Weg, const float* __restrict__ att,
                           float* __restrict__ logits, int E) {
  const int gid  = blockIdx.x * blockDim.x + threadIdx.x;
  const int lane = gid & 31;
  const int wave = gid >> 5;
  const int nw   = (gridDim.x * blockDim.x) >> 5;
  for (int e = wave; e < E; e += nw) {
    const int s = ei[e];
    const int t = ei[E + e];
    const float a0 = ea[e * 5 + 0], a1 = ea[e * 5 + 1], a2 = ea[e * 5 + 2],
                a3 = ea[e * 5 + 3], a4 = ea[e * 5 + 4];
    const float* __restrict__ xls = xl + (long long)s * HD;
    const float* __restrict__ xrt = xr + (long long)t * HD;
#pragma unroll
    for (int h = 0; h < HH; ++h) {
      float part = 0.f;
#pragma unroll
      for (int j = 0; j < 4; ++j) {
        const int d = j * 32 + lane;       // coalesced across the wave
        const int c = h * DD + d;
        const float eg = a0 * Weg[c] + a1 * Weg[HD + c] + a2 * Weg[2 * HD + c] +
                         a3 * Weg[3 * HD + c] + a4 * Weg[4 * HD + c];
        float f = xls[c] + xrt[c] + eg;
        f = (f > 0.f) ? f : 0.2f * f;      // leaky_relu(0.2)
        part += f * att[h * DD + d];
      }
      for (int off = 16; off > 0; off >>= 1) part += __shfl_down(part, off, 32);
      if (lane == 0) logits[e * HH + h] = part;
    }
  }
}

// ---------------- segment max over dst (order-preserving unsigned atomicMax) ----------------
__global__ void seg_max(const float* __restrict__ logits, const int* __restrict__ ei,
                        unsigned* __restrict__ mxu, int E) {
  int i = blockIdx.x * blockDim.x + threadIdx.x;
  const int tot = E * HH;
  const int st = gridDim.x * blockDim.x;
  for (; i < tot; i += st) {
    const int e = i >> 2;
    const int h = i & 3;
    const int t = ei[E + e];
    atomicMax(&mxu[t * HH + h], fenc(logits[i]));
  }
}

// ---------------- exp(logit - max) in place + segment-sum denominator ----------------
__global__ void exp_den(float* __restrict__ logits, const int* __restrict__ ei,
                        const unsigned* __restrict__ mxu, float* __restrict__ den, int E) {
  int i = blockIdx.x * blockDim.x + threadIdx.x;
  const int tot = E * HH;
  const int st = gridDim.x * blockDim.x;
  for (; i < tot; i += st) {
    const int e = i >> 2;
    const int h = i & 3;
    const int t = ei[E + e];
    const float ex = __expf(logits[i] - fdec(mxu[t * HH + h]));
    logits[i] = ex;
    atomicAdd(&den[t * HH + h], ex);
  }
}

// ---------------- weighted scatter, fused head-mean ----------------
__global__ void gat_scatter(const float* __restrict__ xl, const float* __restrict__ ex,
                            const float* __restrict__ den, const int* __restrict__ ei,
                            float* __restrict__ out, int E) {
  int i = blockIdx.x * blockDim.x + threadIdx.x;
  const int tot = E * DD;
  const int st = gridDim.x * blockDim.x;
  for (; i < tot; i += st) {
    const int e = i >> 7;
    const int d = i & 127;
    const int s = ei[e];
    const int t = ei[E + e];
    float a = 0.f;
#pragma unroll
    for (int h = 0; h < HH; ++h)
      a += (ex[e * HH + h] / den[t * HH + h]) * xl[(long long)s * HD + h * DD + d];
    atomicAdd(&out[(long long)t * DD + d], 0.25f * a);
  }
}

// ---------------- final: d_out = (x0 + x1 + BN(x2pre)) / 3 ----------------
__global__ void final_combine(const float* __restrict__ x0, const float* __restrict__ x1,
                              const float* __restrict__ x2pre, const float* __restrict__ mr,
                              const float* __restrict__ g, const float* __restrict__ b,
                              float* __restrict__ out, int M, int N) {
  int i = blockIdx.x * blockDim.x + threadIdx.x;
  const int tot = M * N;
  const int st = gridDim.x * blockDim.x;
  for (; i < tot; i += st) {
    const int c = i % N;
    const float x2 = (x2pre[i] - mr[c]) * mr[N + c] * g[c] + b[c];
    out[i] = (x0[i] + x1[i] + x2) * (1.f / 3.f);
  }
}

// ---------------- host-side orchestration ----------------
static inline int ew_blocks(long long tot) {
  long long b = (tot + 255) / 256;
  if (b > 65536) b = 65536;
  if (b < 1) b = 1;
  return (int)b;
}

static inline void run_bn_stats(const float* X, int M, int N, float* acc, float* mr,
                                hipStream_t s) {
  fill_f32<<<1, 256, 0, s>>>(acc, 0.f, 2 * N);
  dim3 g((N + 255) / 256, 512);
  col_stats<<<g, 256, 0, s>>>(X, M, N, acc);
  finalize_stats<<<(N + 255) / 256, 256, 0, s>>>(acc, N, 1.0f / (float)M, mr);
}

extern "C" void kernel_launch(void* const* d_in, const int* in_sizes, int n_in,
                              void* d_out, int out_size, void* d_ws, size_t ws_size,
                              hipStream_t stream) {
  const float* x       = (const float*)d_in[0];
  const int*   ei      = (const int*)d_in[1];
  const float* ea      = (const float*)d_in[2];
  // d_in[3] = batch (unused: single graph)
  const float* W_in    = (const float*)d_in[4];
  const float* b_in    = (const float*)d_in[5];
  const float* g_in    = (const float*)d_in[6];
  const float* beta_in = (const float*)d_in[7];
  const float* We      = (const float*)d_in[8];
  const float* be      = (const float*)d_in[9];
  const float* W1      = (const float*)d_in[10];
  const float* b1      = (const float*)d_in[11];
  const float* g1      = (const float*)d_in[12];
  const float* beta1   = (const float*)d_in[13];
  const float* W2      = (const float*)d_in[14];
  const float* b2      = (const float*)d_in[15];
  const float* skip0   = (const float*)d_in[16];
  const float* skip1   = (const float*)d_in[17];
  const float* g_n0    = (const float*)d_in[18];
  const float* b_n0    = (const float*)d_in[19];
  const float* g_n1    = (const float*)d_in[20];
  const float* b_n1    = (const float*)d_in[21];
  const float* Wl      = (const float*)d_in[22];
  const float* bl      = (const float*)d_in[23];
  const float* Wr      = (const float*)d_in[24];
  const float* br      = (const float*)d_in[25];
  const float* att     = (const float*)d_in[26];
  const float* W_eg    = (const float*)d_in[27];
  const float* biasgat = (const float*)d_in[28];

  const int Nn = in_sizes[0] / 10;   // nodes
  const int Ee = in_sizes[2] / 5;    // edges

  // ---- workspace layout (floats) ----
  float* ws = (float*)d_ws;
  const size_t ND = (size_t)Nn * DD;
  float* x0   = ws;                 // [N,128]  layer0
  float* x1   = x0 + ND;            // [N,128]  layer1
  float* tmpa = x1 + ND;            // [N,128]  t0 raw / agg / x1pre / x2pre
  float* tmpb = tmpa + ND;          // [N,128]  t2 raw / out_mean
  float* bigA = tmpb + ND;          // [N,512]  t1 / xl
  float* bigB = bigA + 4 * ND;      // [N,512]  xr
  float* elog = bigB + 4 * ND;      // [E,4]    logits -> exp
  float* den  = elog + (size_t)Ee * HH;               // [N,4]
  unsigned* mxu = (unsigned*)(den + (size_t)Nn * HH); // [N,4]
  float* acc  = (float*)(mxu + (size_t)Nn * HH);      // [<=1024] col stats
  float* mr   = acc + 1024;                           // [<=1024] mean/rstd

  const int totND = Nn * DD;

  // ---- Stage A: input projection + BN + ReLU -> x0 ----
  launch_gemm_t<10, 128, false>(x, nullptr, W_in, b_in, tmpa, Nn, stream);
  run_bn_stats(tmpa, Nn, DD, acc, mr, stream);
  bn_apply<<<ew_blocks(totND), 256, 0, stream>>>(tmpa, mr, g_in, beta_in, x0, Nn, DD, 1);

  // ---- Stage B: GINEConv ----
  fill_f32<<<ew_blocks(totND), 256, 0, stream>>>(tmpa, 0.f, totND);            // agg = 0
  gine_edges<<<ew_blocks((long long)Ee * DD), 256, 0, stream>>>(x0, ei, ea, We, be, tmpa, Ee);
  launch_gemm_t<128, 256, true>(x0, tmpa, W1, b1, bigA, Nn, stream);           // (x0+agg)@W1+b1
  run_bn_stats(bigA, Nn, 2 * DD, acc, mr, stream);
  bn_apply<<<ew_blocks((long long)Nn * 2 * DD), 256, 0, stream>>>(bigA, mr, g1, beta1, bigA, Nn, 2 * DD, 1);
  launch_gemm_t<256, 128, false>(bigA, nullptr, W2, b2, tmpb, Nn, stream);     // t2
  skip_mix<<<ew_blocks(totND), 256, 0, stream>>>(x0, tmpb, nullptr, skip0, tmpa, Nn, DD);
  run_bn_stats(tmpa, Nn, DD, acc, mr, stream);
  bn_apply<<<ew_blocks(totND), 256, 0, stream>>>(tmpa, mr, g_n0, b_n0, x1, Nn, DD, 0);

  // ---- Stage C: GATv2Conv ----
  launch_gemm_t<128, 512, false>(x1, nullptr, Wl, bl, bigA, Nn, stream);       // xl
  launch_gemm_t<128, 512, false>(x1, nullptr, Wr, br, bigB, Nn, stream);       // xr
  gat_logits<<<4096, 256, 0, stream>>>(bigA, bigB, ei, ea, W_eg, att, elog, Ee);
  fill_u32<<<ew_blocks((long long)Nn * HH), 256, 0, stream>>>(mxu, 0x007FFFFFu, Nn * HH); // enc(-inf)
  seg_max<<<ew_blocks((long long)Ee * HH), 256, 0, stream>>>(elog, ei, mxu, Ee);
  fill_f32<<<ew_blocks((long long)Nn * HH), 256, 0, stream>>>(den, 0.f, Nn * HH);
  exp_den<<<ew_blocks((long long)Ee * HH), 256, 0, stream>>>(elog, ei, mxu, den, Ee);
  fill_f32<<<ew_blocks(totND), 256, 0, stream>>>(tmpb, 0.f, totND);            // out_mean = 0
  gat_scatter<<<ew_blocks((long long)Ee * DD), 256, 0, stream>>>(bigA, elog, den, ei, tmpb, Ee);
  skip_mix<<<ew_blocks(totND), 256, 0, stream>>>(x1, tmpb, biasgat, skip1, tmpa, Nn, DD);
  run_bn_stats(tmpa, Nn, DD, acc, mr, stream);

  // ---- final: (layer0 + layer1 + BN(x2pre)) / 3 ----
  final_combine<<<ew_blocks(totND), 256, 0, stream>>>(x0, x1, tmpa, mr, g_n1, b_n1,
                                                      (float*)d_out, Nn, DD);
}